// LightGCN_67697274519811
// MI455X (gfx1250) — compile-verified
//
#include <hip/hip_runtime.h>
#include <stdint.h>

// ---------------- problem constants (match reference) ----------------
#define N_USERS 100000
#define N_ITEMS 50000
#define N_NODES_ (N_USERS + N_ITEMS)
#define DIM 64
#define N_HOPS 3
#define EDGE_RATE 0.2f   // keep prob == rate (faithful to torch _sparse_dropout)
#define MESS_KEEP 0.9f   // 1 - mess_dropout_rate

#define EDGE_TILE 256
#define SPMM_MAX_BLOCKS 1536

// Address-space-qualified int ("__device__ int*" / "__shared__ int*" in HIP
// diagnostics) for the gfx1250 async global->LDS builtins.
typedef __attribute__((address_space(1))) int g_i32;
typedef __attribute__((address_space(3))) int l_i32;

#if defined(__has_builtin) && __has_builtin(__builtin_amdgcn_global_load_async_to_lds_b32)
#define HAVE_ASYNC_LDS 1
#else
#define HAVE_ASYNC_LDS 0
#endif

// ---------------- deterministic hash RNG (stand-in for jax PRNG) ------
__device__ __forceinline__ uint32_t hash_u32(uint32_t x) {
  x ^= x >> 16; x *= 0x7feb352dU;
  x ^= x >> 15; x *= 0x846ca68bU;
  x ^= x >> 16;
  return x;
}
__device__ __forceinline__ float u01(uint32_t h) {
  return (float)(h >> 8) * 0x1p-24f;   // [0,1)
}

// ---------------- build all_embed = concat(user, item), float4 -------
__global__ void concat_embed4(const float* __restrict__ ue,
                              const float* __restrict__ ie,
                              float* __restrict__ all_e) {
  const size_t i4  = (size_t)blockIdx.x * blockDim.x + threadIdx.x;
  const size_t nu4 = (size_t)N_USERS * DIM / 4;
  const size_t n4  = (size_t)N_NODES_ * DIM / 4;
  if (i4 < n4) {
    const float4 v = (i4 < nu4) ? ((const float4*)ue)[i4]
                                : ((const float4*)ie)[i4 - nu4];
    ((float4*)all_e)[i4] = v;
  }
}

// ---- stage one tile of edge streams into LDS (async on gfx1250) -----
__device__ __forceinline__ void stage_tile(const int* __restrict__ rows,
                                           const int* __restrict__ cols,
                                           const float* __restrict__ vals,
                                           int nnz, int tile, int t,
                                           int* sr, int* sc, float* sv) {
  const int e = tile * EDGE_TILE + t;
  if (e < nnz) {
#if HAVE_ASYNC_LDS
    const int* gr = rows + e;
    const int* gc = cols + e;
    const int* gv = (const int*)(const void*)(vals + e);
    int* lr = sr + t;
    int* lc = sc + t;
    int* lv = (int*)(void*)(sv + t);
    __builtin_amdgcn_global_load_async_to_lds_b32((g_i32*)gr, (l_i32*)lr, 0, 0);
    __builtin_amdgcn_global_load_async_to_lds_b32((g_i32*)gc, (l_i32*)lc, 0, 0);
    __builtin_amdgcn_global_load_async_to_lds_b32((g_i32*)gv, (l_i32*)lv, 0, 0);
#else
    sr[t] = rows[e];
    sc[t] = cols[e];
    sv[t] = vals[e];
#endif
  }
}

// ---------------- COO SpMM with edge dropout --------------------------
// Grid-stride over 256-edge tiles with a double-buffered async pipeline:
// issue next tile's global->LDS async loads, drain current tile with
// s_wait_asynccnt(3) (async loads retire in order), then 16 lanes/edge
// gather a coalesced 256B row of x (L2-resident) and scatter-add into y
// with f32 atomics.
__global__ __launch_bounds__(EDGE_TILE)
void spmm_edge_dropout(const int* __restrict__ rows,
                       const int* __restrict__ cols,
                       const float* __restrict__ vals,
                       const float* __restrict__ x,
                       float* __restrict__ y,
                       int nnz, int ntiles, uint32_t seed) {
  __shared__ int   s_row[2][EDGE_TILE];
  __shared__ int   s_col[2][EDGE_TILE];
  __shared__ float s_val[2][EDGE_TILE];

  const int t     = threadIdx.x;
  const int chunk = t & 15;   // which float4 of the 64-wide row
  const int grp   = t >> 4;   // which edge inside the 16-edge subtile

  int buf = 0;
  stage_tile(rows, cols, vals, nnz, blockIdx.x, t, s_row[0], s_col[0], s_val[0]);

  for (int tile = blockIdx.x; tile < ntiles; tile += gridDim.x) {
    const int next_tile = tile + (int)gridDim.x;
    if (next_tile < ntiles) {
      stage_tile(rows, cols, vals, nnz, next_tile, t,
                 s_row[buf ^ 1], s_col[buf ^ 1], s_val[buf ^ 1]);
#if HAVE_ASYNC_LDS
      __builtin_amdgcn_s_wait_asynccnt(3);   // only next tile's 3 may remain
#endif
    } else {
#if HAVE_ASYNC_LDS
      __builtin_amdgcn_s_wait_asynccnt(0);
#endif
    }
    __syncthreads();   // current tile's LDS data visible to all waves

    const int base = tile * EDGE_TILE;
    for (int s = 0; s < EDGE_TILE; s += 16) {
      const int ei = s + grp;
      const int ge = base + ei;
      if (ge < nnz) {
        const int   r = s_row[buf][ei];
        const int   c = s_col[buf][ei];
        float       v = s_val[buf][ei];
        // keep where floor(rate + u) == 1  <=>  u >= 1 - rate (keep prob = rate)
        const float u = u01(hash_u32(seed ^ (uint32_t)ge));
        if (EDGE_RATE + u >= 1.0f) {
          v *= 1.0f / (1.0f - EDGE_RATE);
          const float4 xv = *(const float4*)(x + (size_t)c * DIM + (size_t)chunk * 4);
          float* yp = y + (size_t)r * DIM + (size_t)chunk * 4;
          atomicAdd(yp + 0, xv.x * v);
          atomicAdd(yp + 1, xv.y * v);
          atomicAdd(yp + 2, xv.z * v);
          atomicAdd(yp + 3, xv.w * v);
        }
      }
    }

    __syncthreads();   // all reads done before this buffer is overwritten
    buf ^= 1;
  }
}

// ------- message dropout (in place) + accumulate into sum, float4 -----
__global__ void scale_accum4(float* __restrict__ buf,
                             float* __restrict__ sum,
                             uint32_t seed, size_t n4) {
  const size_t i4 = (size_t)blockIdx.x * blockDim.x + threadIdx.x;
  if (i4 >= n4) return;
  float4 v = ((float4*)buf)[i4];
  const uint32_t b = (uint32_t)(i4 * 4);
  float* vp = (float*)&v;
#pragma unroll
  for (int k = 0; k < 4; ++k) {
    const float u = u01(hash_u32(seed ^ (b + k)));
    vp[k] = (u < MESS_KEEP) ? vp[k] * (1.0f / MESS_KEEP) : 0.0f;
  }
  ((float4*)buf)[i4] = v;            // becomes `cur` for the next hop
  float4 s = ((float4*)sum)[i4];
  s.x += v.x; s.y += v.y; s.z += v.z; s.w += v.w;
  ((float4*)sum)[i4] = s;
}

// ------- mean over hops, write into concatenated output, float4 -------
__global__ void finalize_mean4(const float* __restrict__ sum,
                               float* __restrict__ out, int rel) {
  const size_t i4 = (size_t)blockIdx.x * blockDim.x + threadIdx.x;
  const size_t n4 = (size_t)N_NODES_ * DIM / 4;
  if (i4 >= n4) return;
  const int n  = (int)(i4 >> 4);      // node id (16 float4 per 64-wide row)
  const int d4 = (int)(i4 & 15);      // float4 index inside the row
  float4 v = ((const float4*)sum)[i4];
  const float sc = 1.0f / (N_HOPS + 1);
  v.x *= sc; v.y *= sc; v.z *= sc; v.w *= sc;
  // output row = 3*DIM floats = 48 float4; relation segment = 16 float4
  size_t o4;
  if (n < N_USERS) {
    o4 = (size_t)n * 48 + (size_t)rel * 16 + d4;
  } else {
    o4 = (size_t)N_USERS * 48 + (size_t)(n - N_USERS) * 48 +
         (size_t)rel * 16 + d4;
  }
  ((float4*)out)[o4] = v;
}

// ----------------------------- driver ---------------------------------
extern "C" void kernel_launch(void* const* d_in, const int* in_sizes, int n_in,
                              void* d_out, int out_size, void* d_ws, size_t ws_size,
                              hipStream_t stream) {
  (void)n_in; (void)out_size; (void)ws_size;

  const float* ue = (const float*)d_in[0];
  const float* ie = (const float*)d_in[1];
  const int*   rows[3] = {(const int*)d_in[2], (const int*)d_in[5], (const int*)d_in[8]};
  const int*   cols[3] = {(const int*)d_in[3], (const int*)d_in[6], (const int*)d_in[9]};
  const float* vals[3] = {(const float*)d_in[4], (const float*)d_in[7], (const float*)d_in[10]};
  float* out = (float*)d_out;

  const size_t NE  = (size_t)N_NODES_ * DIM;    // 9.6M floats per buffer
  const size_t NE4 = NE / 4;
  float* all_e = (float*)d_ws;
  float* sum   = all_e + NE;
  float* bufA  = sum + NE;
  float* bufB  = bufA + NE;

  const int TB = 256;
  const int nblk4 = (int)((NE4 + TB - 1) / TB);

  concat_embed4<<<nblk4, TB, 0, stream>>>(ue, ie, all_e);

  for (int r = 0; r < 3; ++r) {
    const int nnz = in_sizes[2 + r * 3];
    const int ntiles = (nnz + EDGE_TILE - 1) / EDGE_TILE;
    const int eblk = ntiles < SPMM_MAX_BLOCKS ? ntiles : SPMM_MAX_BLOCKS;

    // sum starts as hop-0 embedding
    (void)hipMemcpyAsync(sum, all_e, NE * sizeof(float),
                         hipMemcpyDeviceToDevice, stream);

    const float* cur = all_e;
    float* nxt = bufA;
    for (int h = 1; h <= N_HOPS; ++h) {
      (void)hipMemsetAsync(nxt, 0, NE * sizeof(float), stream);

      const uint32_t eseed = 0x9E3779B9u * (uint32_t)(r * 8 + h) + 0x85ebca6bu;
      spmm_edge_dropout<<<eblk, EDGE_TILE, 0, stream>>>(
          rows[r], cols[r], vals[r], cur, nxt, nnz, ntiles, eseed);

      const uint32_t mseed = 0xC2B2AE35u * (uint32_t)(r * 8 + h) + 0x27d4eb2fu;
      scale_accum4<<<nblk4, TB, 0, stream>>>(nxt, sum, mseed, NE4);

      cur = nxt;
      nxt = (nxt == bufA) ? bufB : bufA;
    }

    finalize_mean4<<<nblk4, TB, 0, stream>>>(sum, out, r);
  }
}